// LSTMpool_47132971106335
// MI455X (gfx1250) — compile-verified
//
#include <hip/hip_runtime.h>
#include <hip/hip_bf16.h>
#include <math.h>

// Problem constants (from reference)
#define VOCAB 50000
#define EMB   300
#define HID   16
#define GATES 64      // 4*HID
#define OUTD  5
#define BATCH 128
#define SEQ   512

typedef __attribute__((ext_vector_type(2))) float v2f;
typedef __attribute__((ext_vector_type(8))) float v8f;

// ---------------------------------------------------------------------------
// Kernel 1: P[v, j] = sum_k emb_table[v,k] * w_ih[j,k]   (v<50000, j<64)
// Full-f32 WMMA: V_WMMA_F32_16X16X4_F32, K looped 300/4 = 75 chunks.
// One wave per 16-row M strip; 4 N-tiles (N=64) accumulated in registers.
// ---------------------------------------------------------------------------
#define MTILES ((VOCAB + 15) / 16)   // 3125 (exact: 50000/16)
#define WAVES_PER_BLOCK 4

__global__ void __launch_bounds__(32 * WAVES_PER_BLOCK)
emb_proj_gemm(const float* __restrict__ emb,
              const float* __restrict__ w_ih,
              float* __restrict__ P)
{
    const int wave  = threadIdx.x >> 5;
    const int lane  = threadIdx.x & 31;
    const int mtile = blockIdx.x * WAVES_PER_BLOCK + wave;
    if (mtile >= MTILES) return;

    const int mrow  = mtile * 16 + (lane & 15);    // A: M = lane%16
    const int khalf = (lane >> 4) * 2;             // A/B: K = khalf + {0,1}
    const int ncol  = (lane & 15);                 // B: N = lane%16

    const float* __restrict__ arow = emb + (size_t)mrow * EMB + khalf;

    v8f acc0 = {}; v8f acc1 = {}; v8f acc2 = {}; v8f acc3 = {};

    for (int k0 = 0; k0 < EMB; k0 += 4) {
        v2f a = *(const v2f*)(arow + k0);   // 8B aligned: even float offset

        const float* b0 = w_ih + (size_t)(0*16 + ncol) * EMB + khalf + k0;
        const float* b1 = w_ih + (size_t)(1*16 + ncol) * EMB + khalf + k0;
        const float* b2 = w_ih + (size_t)(2*16 + ncol) * EMB + khalf + k0;
        const float* b3 = w_ih + (size_t)(3*16 + ncol) * EMB + khalf + k0;
        v2f bv0 = *(const v2f*)b0;
        v2f bv1 = *(const v2f*)b1;
        v2f bv2 = *(const v2f*)b2;
        v2f bv3 = *(const v2f*)b3;

        acc0 = __builtin_amdgcn_wmma_f32_16x16x4_f32(false, a, false, bv0,
                                                     (short)0, acc0, false, false);
        acc1 = __builtin_amdgcn_wmma_f32_16x16x4_f32(false, a, false, bv1,
                                                     (short)0, acc1, false, false);
        acc2 = __builtin_amdgcn_wmma_f32_16x16x4_f32(false, a, false, bv2,
                                                     (short)0, acc2, false, false);
        acc3 = __builtin_amdgcn_wmma_f32_16x16x4_f32(false, a, false, bv3,
                                                     (short)0, acc3, false, false);
    }

    // C/D layout: lane<16 -> M=r, N=lane ; lane>=16 -> M=r+8, N=lane-16
    const int Mbase = (lane < 16) ? 0 : 8;
#pragma unroll
    for (int r = 0; r < 8; ++r) {
        const size_t row = (size_t)(mtile * 16 + Mbase + r) * GATES;
        P[row + 0*16 + ncol] = acc0[r];
        P[row + 1*16 + ncol] = acc1[r];
        P[row + 2*16 + ncol] = acc2[r];
        P[row + 3*16 + ncol] = acc3[r];
    }
}

// ---------------------------------------------------------------------------
// Kernel 2: sequential LSTM recurrence + pooling + 48->5 linear.
// One wave32 per batch element. Lane j owns gate rows j and j+32 of w_hh.
// Gate order (torch): i=[0,16), f=[16,32), g=[32,48), o=[48,64).
//   lane u (<16)  computes (i_u, g_u);  lane u+16 computes (f_u, o_u).
// ---------------------------------------------------------------------------
__device__ __forceinline__ float sigm(float v) {
    return 1.0f / (1.0f + __expf(-v));
}

__global__ void __launch_bounds__(32)
lstm_recurrence(const int*   __restrict__ x,
                const int*   __restrict__ lengths,
                const float* __restrict__ P,
                const float* __restrict__ b_ih,
                const float* __restrict__ b_hh,
                const float* __restrict__ w_hh,
                const float* __restrict__ w_lin,
                const float* __restrict__ b_lin,
                float*       __restrict__ out)
{
    const int b = blockIdx.x;
    const int l = threadIdx.x;                 // 0..31

    __shared__ float h_lds[HID];
    __shared__ float rep[3 * HID];

    // Recurrent weights for the two gate rows this lane owns.
    float w0[HID], w1[HID];
#pragma unroll
    for (int k = 0; k < HID; ++k) {
        w0[k] = w_hh[(size_t)l * HID + k];
        w1[k] = w_hh[(size_t)(l + 32) * HID + k];
    }
    const float bias0 = b_ih[l]      + b_hh[l];
    const float bias1 = b_ih[l + 32] + b_hh[l + 32];

    if (l < HID) h_lds[l] = 0.0f;
    float c = 0.0f, hval = 0.0f;
    float psum = 0.0f, pmax = -INFINITY, plast = 0.0f;

    int L = lengths[b];
    L = (L < 1) ? 1 : ((L > SEQ) ? SEQ : L);

    const int* __restrict__ xb = x + (size_t)b * SEQ;
    __syncthreads();

    for (int t = 0; t < L; ++t) {
        const int idx = xb[t];
        const float* __restrict__ prow = P + (size_t)idx * GATES;
        float acc0 = prow[l]      + bias0;
        float acc1 = prow[l + 32] + bias1;

        // Split accumulation halves the dependent FMA chain.
        float s0a = 0.f, s0b = 0.f, s1a = 0.f, s1b = 0.f;
#pragma unroll
        for (int k = 0; k < HID; k += 2) {
            const float hk0 = h_lds[k];        // broadcast read: conflict-free
            const float hk1 = h_lds[k + 1];
            s0a = fmaf(w0[k],     hk0, s0a);
            s0b = fmaf(w0[k + 1], hk1, s0b);
            s1a = fmaf(w1[k],     hk0, s1a);
            s1b = fmaf(w1[k + 1], hk1, s1b);
        }
        acc0 += s0a + s0b;
        acc1 += s1a + s1b;

        // lane<16: (i, g)  ;  lane>=16: (f, o)
        const float a0 = sigm(acc0);
        const float a1 = (l < HID) ? tanhf(acc1) : sigm(acc1);
        const float fo0 = __shfl_xor(a0, 16);  // lane u<16 receives f_u
        const float fo1 = __shfl_xor(a1, 16);  // lane u<16 receives o_u

        if (l < HID) {
            c    = fo0 * c + a0 * a1;          // c = f*c + i*g
            hval = fo1 * tanhf(c);             // h = o*tanh(c)
            h_lds[l] = hval;
            psum += hval;
            pmax  = fmaxf(pmax, hval);
            if (t == L - 1) plast = hval;
        }
        __syncthreads();                       // single wave: ~free, keeps LDS ordered
    }

    // rep = [last(16) | sum(16) | max(16)]
    if (l < HID) {
        rep[l]           = plast;
        rep[HID + l]     = psum;
        rep[2 * HID + l] = pmax;
    }
    __syncthreads();

    if (l < OUTD) {
        float acc = b_lin[l];
#pragma unroll
        for (int j = 0; j < 3 * HID; ++j)
            acc = fmaf(w_lin[(size_t)l * (3 * HID) + j], rep[j], acc);
        out[(size_t)b * OUTD + l] = acc;
    }
}

// ---------------------------------------------------------------------------
// Launch
// ---------------------------------------------------------------------------
extern "C" void kernel_launch(void* const* d_in, const int* in_sizes, int n_in,
                              void* d_out, int out_size, void* d_ws, size_t ws_size,
                              hipStream_t stream)
{
    const int*   x      = (const int*)  d_in[0];
    const int*   len    = (const int*)  d_in[1];
    const float* emb    = (const float*)d_in[2];
    const float* w_ih   = (const float*)d_in[3];
    const float* w_hh   = (const float*)d_in[4];
    const float* b_ih   = (const float*)d_in[5];
    const float* b_hh   = (const float*)d_in[6];
    const float* w_lin  = (const float*)d_in[7];
    const float* b_lin  = (const float*)d_in[8];
    float*       out    = (float*)d_out;

    // Projected embedding table P[VOCAB, 64] in workspace (12.8 MB, L2-resident).
    float* P = (float*)d_ws;

    const int blocks1 = (MTILES + WAVES_PER_BLOCK - 1) / WAVES_PER_BLOCK;
    emb_proj_gemm<<<blocks1, 32 * WAVES_PER_BLOCK, 0, stream>>>(emb, w_ih, P);
    lstm_recurrence<<<BATCH, 32, 0, stream>>>(x, len, P, b_ih, b_hh,
                                              w_hh, w_lin, b_lin, out);
}